// EdgeConvModule_33492154974284
// MI455X (gfx1250) — compile-verified
//
#include <hip/hip_runtime.h>
#include <hip/hip_bf16.h>

typedef __attribute__((ext_vector_type(2))) float v2f;
typedef __attribute__((ext_vector_type(8))) float v8f;

#define B_    8
#define C_    64
#define N_    4096
#define OUT_  64
#define K_    20
#define C2_   128

__device__ __forceinline__ v8f wmma_f32(v2f a, v2f b, v8f c) {
    return __builtin_amdgcn_wmma_f32_16x16x4_f32(false, a, false, b, (short)0, c,
                                                 false, false);
}

// ---------------------------------------------------------------- init accum
__global__ __launch_bounds__(256) void init_acc_kernel(float* __restrict__ acc) {
    int i = blockIdx.x * 256 + threadIdx.x;
    if (i < 64 * 128) acc[i] = 0.0f;
}

// ---------------------------------------------------------------- sq norms
__global__ __launch_bounds__(256) void norms_kernel(const float* __restrict__ X,
                                                    float* __restrict__ sq) {
    int gid = blockIdx.x * 256 + threadIdx.x;     // B_*N_ threads
    int b = gid >> 12;                            // /N_
    int n = gid & (N_ - 1);
    const float* Xb = X + (size_t)b * C_ * N_;
    float s = 0.0f;
#pragma unroll
    for (int c = 0; c < C_; ++c) {
        float v = Xb[c * N_ + n];
        s += v * v;
    }
    sq[gid] = s;
}

// ---------------------------------------------------------------- KNN (WMMA)
// One wave per 32 rows. pd[n][m] = 2*dot(x_n,x_m) - sq_n - sq_m ; top-20 per row.
__global__ __launch_bounds__(32) void knn_kernel(const float* __restrict__ X,
                                                 const float* __restrict__ sq,
                                                 int* __restrict__ idx_out) {
    const int b    = blockIdx.y;
    const int n0   = blockIdx.x * 32;
    const int lane = threadIdx.x;
    const float* Xb  = X + (size_t)b * C_ * N_;
    const float* sqb = sq + b * N_;

    __shared__ float pdl[32][17];

    // A fragments for two 16-row tiles (16x64 each), fp32 16x16x4 layout:
    // lane holds row M=lane&15, channels 4f + 2*(lane>>4) + {0,1}
    v2f a0[16], a1[16];
    {
        int row    = lane & 15;
        int chalf  = (lane >> 4) * 2;
#pragma unroll
        for (int f = 0; f < 16; ++f) {
            int c = 4 * f + chalf;
            a0[f].x = Xb[c * N_ + n0 + row];
            a0[f].y = Xb[(c + 1) * N_ + n0 + row];
            a1[f].x = Xb[c * N_ + n0 + 16 + row];
            a1[f].y = Xb[(c + 1) * N_ + n0 + 16 + row];
        }
    }
    const float sq_own = sqb[n0 + lane];

    float tv[20];
    int   ti[20];
#pragma unroll
    for (int j = 0; j < 20; ++j) { tv[j] = -3.0e38f; ti[j] = 0; }

    for (int m0 = 0; m0 < N_; m0 += 16) {
        v8f acc0 = {}; v8f acc1 = {};
        const int col   = m0 + (lane & 15);
        const int chalf = (lane >> 4) * 2;
#pragma unroll
        for (int f = 0; f < 16; ++f) {
            int c = 4 * f + chalf;
            v2f bf;
            bf.x = Xb[c * N_ + col];
            bf.y = Xb[(c + 1) * N_ + col];
            acc0 = wmma_f32(a0[f], bf, acc0);
            acc1 = wmma_f32(a1[f], bf, acc1);
        }
        // C layout: lane holds N=lane&15, rows M = v + 8*(lane>>4)
        const float sqm   = sqb[col];
        const int   rbase = 8 * (lane >> 4);
        const int   ccol  = lane & 15;
#pragma unroll
        for (int v = 0; v < 8; ++v) {
            pdl[rbase + v][ccol]      = 2.0f * acc0[v] - sqm;
            pdl[16 + rbase + v][ccol] = 2.0f * acc1[v] - sqm;
        }
        __syncthreads();
        // each lane owns row n0+lane: scan 16 candidates, maintain sorted top-20
#pragma unroll
        for (int j = 0; j < 16; ++j) {
            float val = pdl[lane][j] - sq_own;
            if (val > tv[19]) {
                tv[19] = val; ti[19] = m0 + j;
#pragma unroll
                for (int s = 19; s > 0; --s) {
                    if (tv[s] > tv[s - 1]) {
                        float t = tv[s]; tv[s] = tv[s - 1]; tv[s - 1] = t;
                        int   u = ti[s]; ti[s] = ti[s - 1]; ti[s - 1] = u;
                    }
                }
            }
        }
        __syncthreads();
    }
    int* op = idx_out + ((size_t)b * N_ + n0 + lane) * K_;
#pragma unroll
    for (int j = 0; j < 20; ++j) op[j] = ti[j];
}

// ---------------------------------------------------------------- stats GEMM
// 16 edge rows per wave; y = edge(128) x W^T(128->64); accumulate sum/sumsq
// per output channel into one of 64 slice accumulators.
__global__ __launch_bounds__(32) void stats_kernel(const float* __restrict__ X,
                                                   const float* __restrict__ W,
                                                   const int* __restrict__ idx,
                                                   float* __restrict__ acc) {
    const int lane = threadIdx.x;
    const int r    = blockIdx.x * 16 + (lane & 15);   // this lane's A row
    const int b    = r / (N_ * K_);
    const int rem  = r - b * (N_ * K_);
    const int n    = rem / K_;
    const int nb   = idx[r];
    const float* Xb = X + (size_t)b * C_ * N_;
    const int chalf = (lane >> 4) * 2;

    // edge fragments: [0..63] = neighbor - center, [64..127] = center
    v2f a[32];
#pragma unroll
    for (int f = 0; f < 16; ++f) {
        int c = 4 * f + chalf;
        v2f cen, nei;
        cen.x = Xb[c * N_ + n];        cen.y = Xb[(c + 1) * N_ + n];
        nei.x = Xb[c * N_ + nb];       nei.y = Xb[(c + 1) * N_ + nb];
        a[f]      = nei - cen;
        a[f + 16] = cen;
    }

    const int slice = blockIdx.x & 63;
    float* accS = acc + slice * 128;

    for (int ot = 0; ot < 4; ++ot) {
        v8f yacc = {};
        const int o = ot * 16 + (lane & 15);
#pragma unroll
        for (int f = 0; f < 32; ++f) {
            int c2 = 4 * f + chalf;
            v2f bf;
            bf.x = W[o * C2_ + c2];
            bf.y = W[o * C2_ + c2 + 1];
            yacc = wmma_f32(a[f], bf, yacc);
        }
        float s = 0.0f, q = 0.0f;
#pragma unroll
        for (int v = 0; v < 8; ++v) { float y = yacc[v]; s += y; q += y * y; }
        s += __shfl_xor(s, 16);
        q += __shfl_xor(q, 16);
        if (lane < 16) {
            int ch = ot * 16 + lane;
            atomicAdd(&accS[ch], s);
            atomicAdd(&accS[64 + ch], q);
        }
    }
}

// ---------------------------------------------------------------- finalize BN
__global__ __launch_bounds__(64) void finalize_kernel(const float* __restrict__ acc,
                                                      const float* __restrict__ gamma,
                                                      const float* __restrict__ beta,
                                                      float* __restrict__ sclsft) {
    int ch = threadIdx.x;
    float s = 0.0f, q = 0.0f;
    for (int sl = 0; sl < 64; ++sl) {
        s += acc[sl * 128 + ch];
        q += acc[sl * 128 + 64 + ch];
    }
    const float cnt  = (float)(B_ * N_ * K_);
    float mean = s / cnt;
    float var  = q / cnt - mean * mean;
    float inv  = rsqrtf(var + 1e-5f);
    float scl  = gamma[ch] * inv;
    sclsft[ch]      = scl;
    sclsft[64 + ch] = beta[ch] - mean * scl;
}

// ---------------------------------------------------------------- output pass
// One wave per (b, n): recompute y for k=0..19 (two 16-row tiles, clamped),
// normalize + leaky-relu, max over k, write out[b][o][n].
__global__ __launch_bounds__(32) void output_kernel(const float* __restrict__ X,
                                                    const float* __restrict__ W,
                                                    const int* __restrict__ idx,
                                                    const float* __restrict__ sclsft,
                                                    float* __restrict__ out) {
    const int b    = blockIdx.y;
    const int n    = blockIdx.x;
    const int lane = threadIdx.x;
    const float* Xb = X + (size_t)b * C_ * N_;
    const int* ib   = idx + ((size_t)b * N_ + n) * K_;
    const int chalf = (lane >> 4) * 2;

    const int k0  = lane & 15;               // tile0 row
    const int k1  = min(16 + (lane & 15), K_ - 1); // tile1 row (clamped dup)
    const int nb0 = ib[k0];
    const int nb1 = ib[k1];

    v2f cen[16], d0[16], d1[16];
#pragma unroll
    for (int f = 0; f < 16; ++f) {
        int c = 4 * f + chalf;
        v2f c2v, e0, e1;
        c2v.x = Xb[c * N_ + n];    c2v.y = Xb[(c + 1) * N_ + n];
        e0.x  = Xb[c * N_ + nb0];  e0.y  = Xb[(c + 1) * N_ + nb0];
        e1.x  = Xb[c * N_ + nb1];  e1.y  = Xb[(c + 1) * N_ + nb1];
        cen[f] = c2v;
        d0[f]  = e0 - c2v;
        d1[f]  = e1 - c2v;
    }

    float* ob = out + (size_t)b * OUT_ * N_;

    for (int ot = 0; ot < 4; ++ot) {
        v8f acc0 = {}; v8f acc1 = {};
        const int o = ot * 16 + (lane & 15);
#pragma unroll
        for (int f = 0; f < 32; ++f) {
            int c2 = 4 * f + chalf;
            v2f bf;
            bf.x = W[o * C2_ + c2];
            bf.y = W[o * C2_ + c2 + 1];
            v2f af0 = (f < 16) ? d0[f & 15] : cen[f & 15];
            v2f af1 = (f < 16) ? d1[f & 15] : cen[f & 15];
            acc0 = wmma_f32(af0, bf, acc0);
            acc1 = wmma_f32(af1, bf, acc1);
        }
        const float scl = sclsft[o];
        const float sft = sclsft[64 + o];
        float m = -3.0e38f;
#pragma unroll
        for (int v = 0; v < 8; ++v) {
            float y = acc0[v] * scl + sft;            // rows k = v + 8*(lane>>4), all valid
            y = (y >= 0.0f) ? y : 0.2f * y;
            m = fmaxf(m, y);
        }
        const bool lowhalf = (lane < 16);
#pragma unroll
        for (int v = 0; v < 4; ++v) {                 // rows k = 16+v (low half only)
            float y = acc1[v] * scl + sft;
            y = (y >= 0.0f) ? y : 0.2f * y;
            m = (lowhalf) ? fmaxf(m, y) : m;
        }
        m = fmaxf(m, __shfl_xor(m, 16));              // combine lane halves (same channel)
        if (lane < 16) {
            int ch = ot * 16 + lane;
            ob[ch * N_ + n] = m;
        }
    }
}

// ---------------------------------------------------------------- launcher
extern "C" void kernel_launch(void* const* d_in, const int* in_sizes, int n_in,
                              void* d_out, int out_size, void* d_ws, size_t ws_size,
                              hipStream_t stream) {
    const float* X     = (const float*)d_in[0];  // (B, C, N)
    const float* W     = (const float*)d_in[1];  // (OUT, 2C)
    const float* gamma = (const float*)d_in[2];  // (OUT,)
    const float* beta  = (const float*)d_in[3];  // (OUT,)
    float* out = (float*)d_out;                  // (B, OUT, N)

    float* acc    = (float*)d_ws;                // 64 slices * 128 f32
    float* sclsft = acc + 64 * 128;              // 128 f32
    float* sq     = sclsft + 128;                // B*N f32
    int*   idx    = (int*)(sq + B_ * N_);        // B*N*K int

    init_acc_kernel<<<dim3((64 * 128 + 255) / 256), dim3(256), 0, stream>>>(acc);
    norms_kernel<<<dim3(B_ * N_ / 256), dim3(256), 0, stream>>>(X, sq);
    knn_kernel<<<dim3(N_ / 32, B_), dim3(32), 0, stream>>>(X, sq, idx);
    stats_kernel<<<dim3(B_ * N_ * K_ / 16), dim3(32), 0, stream>>>(X, W, idx, acc);
    finalize_kernel<<<dim3(1), dim3(64), 0, stream>>>(acc, gamma, beta, sclsft);
    output_kernel<<<dim3(N_, B_), dim3(32), 0, stream>>>(X, W, idx, sclsft, out);
}